// MultiAggLP_no_macro_18159121728003
// MI455X (gfx1250) — compile-verified
//
#include <hip/hip_runtime.h>
#include <math.h>

typedef __attribute__((ext_vector_type(2))) float v2f;
typedef __attribute__((ext_vector_type(8))) float v8f;

#define TPB 256

static inline int cdiv(long long a, long long b) { return (int)((a + b - 1) / b); }

__device__ __forceinline__ float sigmoidf_(float x) { return 1.0f / (1.0f + expf(-x)); }

__device__ __forceinline__ void atomicMaxF(float* addr, float val) {
  unsigned int* ua = (unsigned int*)addr;
  unsigned int old = __float_as_uint(*addr);
  while (__uint_as_float(old) < val) {
    unsigned int assumed = old;
    old = atomicCAS(ua, assumed, __float_as_uint(val));
    if (old == assumed) break;
  }
}

__global__ void fill_kernel(float* p, float v, int n) {
  int i = blockIdx.x * blockDim.x + threadIdx.x;
  if (i < n) p[i] = v;
}

// C[M,Nc] = A[M,K] @ B[K,Nc] (+bias) ; acc=1 -> C += result. All dims: M%16==0, Nc%16==0, K%4==0.
// One wave per 16x16 tile, V_WMMA_F32_16X16X4_F32, fp32 accumulate (matches reference numerics).
__global__ void gemm_wmma_kernel(const float* __restrict__ A, const float* __restrict__ B,
                                 const float* __restrict__ bias, float* __restrict__ C,
                                 int M, int K, int Nc, int acc) {
  int wave = (int)(((unsigned)blockIdx.x * blockDim.x + threadIdx.x) >> 5);
  int lane = threadIdx.x & 31;
  int tilesN = Nc >> 4;
  int total = (M >> 4) * tilesN;
  if (wave >= total) return;  // whole-wave exit: EXEC stays all-ones for WMMA
  int i0 = (wave / tilesN) << 4;
  int j0 = (wave % tilesN) << 4;
  int rA = i0 + (lane & 15);
  int cB = j0 + (lane & 15);
  int kOff = (lane >> 4) << 1;  // lanes 0-15: K {0,1}; lanes 16-31: K {2,3}
  v8f c = {};
  for (int k = 0; k < K; k += 4) {
    const float* ap = A + (size_t)rA * K + (k + kOff);
    const float* bp = B + (size_t)(k + kOff) * Nc + cB;
    v2f a, b;
    a.x = ap[0]; a.y = ap[1];
    b.x = bp[0]; b.y = bp[Nc];
    c = __builtin_amdgcn_wmma_f32_16x16x4_f32(false, a, false, b, (short)0, c, false, false);
  }
  float bv = bias ? bias[cB] : 0.0f;
  int mBase = i0 + ((lane >> 4) << 3);
#pragma unroll
  for (int r = 0; r < 8; ++r) {
    size_t o = (size_t)(mBase + r) * Nc + cB;
    float v = c[r] + bv;
    C[o] = acc ? (C[o] + v) : v;
  }
}

// Per-node attention dots: alpha_src[n,h] = sum_d h[n,h,d]*a_s[h,d] (same for dst)
__global__ void node_alpha_kernel(const float* __restrict__ hW, const float* __restrict__ a_s,
                                  const float* __restrict__ a_d, float* __restrict__ alS,
                                  float* __restrict__ alD, int Nn, int H, int dh) {
  int idx = blockIdx.x * blockDim.x + threadIdx.x;
  if (idx >= Nn * H) return;
  int n = idx / H, h = idx % H;
  const float* hp = hW + (size_t)n * H * dh + h * dh;
  float s = 0.f, d = 0.f;
  for (int j = 0; j < dh; ++j) { s += hp[j] * a_s[h * dh + j]; d += hp[j] * a_d[h * dh + j]; }
  alS[idx] = s; alD[idx] = d;
}

__global__ void edge_max_kernel(const int* __restrict__ src, const int* __restrict__ dst,
                                const float* __restrict__ alS, const float* __restrict__ alD,
                                float* __restrict__ m, int E, int H) {
  int idx = blockIdx.x * blockDim.x + threadIdx.x;
  if (idx >= E * H) return;
  int e = idx / H, h = idx % H;
  float v = alS[src[e] * H + h] + alD[dst[e] * H + h];
  v = v > 0.f ? v : 0.2f * v;  // leaky_relu(0.2)
  atomicMaxF(&m[dst[e] * H + h], v);
}

__global__ void fix_m_kernel(float* m, int n) {
  int i = blockIdx.x * blockDim.x + threadIdx.x;
  if (i < n) { float v = m[i]; if (!isfinite(v)) m[i] = 0.f; }
}

__global__ void edge_accum_kernel(const int* __restrict__ src, const int* __restrict__ dst,
                                  const float* __restrict__ alS, const float* __restrict__ alD,
                                  const float* __restrict__ m, const float* __restrict__ ew,
                                  const float* __restrict__ hW, float* __restrict__ den,
                                  float* __restrict__ num, int E, int H, int dh) {
  int idx = blockIdx.x * blockDim.x + threadIdx.x;
  if (idx >= E * H) return;
  int e = idx / H, h = idx % H;
  int s = src[e], d = dst[e];
  float v = alS[s * H + h] + alD[d * H + h];
  v = v > 0.f ? v : 0.2f * v;
  float ex = expf(v - m[d * H + h]) * ew[e];
  atomicAdd(&den[d * H + h], ex);
  const float* hp = hW + (size_t)s * H * dh + h * dh;
  float* np = num + (size_t)d * H * dh + h * dh;
  for (int j = 0; j < dh; ++j) atomicAdd(&np[j], ex * hp[j]);
}

__global__ void gat_out_kernel(const float* __restrict__ num, const float* __restrict__ den,
                               float* __restrict__ outp, int Nn, int H, int dh) {
  int Fo = H * dh;
  int idx = blockIdx.x * blockDim.x + threadIdx.x;
  if (idx >= Nn * Fo) return;
  int n = idx / Fo, f = idx % Fo;
  float v = num[idx] / (den[n * H + f / dh] + 1e-16f);
  outp[idx] = v > 0.f ? v : (expf(v) - 1.f);  // elu
}

__global__ void meso_kernel(const float* __restrict__ g2, const int* __restrict__ part,
                            const float* __restrict__ Dw, float* __restrict__ msum,
                            float* __restrict__ mden, int Nn, int dd) {
  int idx = blockIdx.x * blockDim.x + threadIdx.x;
  if (idx >= Nn * dd) return;
  int n = idx / dd, j = idx % dd;
  int p = part[n];
  atomicAdd(&msum[p * dd + j], Dw[n] * g2[idx]);
  if (j == 0) atomicAdd(&mden[p], Dw[n]);
}

__global__ void agg_kernel(float* __restrict__ agg, const float* __restrict__ g2,
                           const float* __restrict__ msum, const float* __restrict__ mden,
                           const int* __restrict__ part, int Nn, int dd) {
  int idx = blockIdx.x * blockDim.x + threadIdx.x;
  if (idx >= Nn * dd) return;
  int n = idx / dd, j = idx % dd;
  int p = part[n];
  agg[(size_t)n * 2 * dd + j] = g2[idx];
  agg[(size_t)n * 2 * dd + dd + j] = msum[p * dd + j] / (mden[p] + 1e-16f);
}

__global__ void lstm_kernel(const float* __restrict__ z, float* __restrict__ h,
                            float* __restrict__ c, int Nn, int dh) {
  int idx = blockIdx.x * blockDim.x + threadIdx.x;
  if (idx >= Nn * dh) return;
  int n = idx / dh, j = idx % dh;
  const float* zr = z + (size_t)n * 4 * dh;
  float ig = sigmoidf_(zr[j]);
  float fg = sigmoidf_(zr[dh + j]);
  float gg = tanhf(zr[2 * dh + j]);
  float og = sigmoidf_(zr[3 * dh + j]);
  float cn = fg * c[idx] + ig * gg;
  c[idx] = cn;
  h[idx] = og * tanhf(cn);
}

__global__ void tanh_kernel(float* p, int n) {
  int i = blockIdx.x * blockDim.x + threadIdx.x;
  if (i < n) p[i] = tanhf(p[i]);
}
__global__ void sigmoid_kernel(float* p, int n) {
  int i = blockIdx.x * blockDim.x + threadIdx.x;
  if (i < n) p[i] = sigmoidf_(p[i]);
}
__global__ void colsq_kernel(const float* __restrict__ emb, float* __restrict__ acc, int n, int dd) {
  int i = blockIdx.x * blockDim.x + threadIdx.x;
  if (i < n) atomicAdd(&acc[i % dd], emb[i] * emb[i]);
}
__global__ void colscale_kernel(float* emb, const float* __restrict__ acc, int n, int dd) {
  int i = blockIdx.x * blockDim.x + threadIdx.x;
  if (i < n) emb[i] /= sqrtf(acc[i % dd]);
}
__global__ void sqnorm_kernel(const float* __restrict__ emb, float* __restrict__ sq, int Nn, int dd) {
  int n = blockIdx.x * blockDim.x + threadIdx.x;
  if (n >= Nn) return;
  const float* p = emb + (size_t)n * dd;
  float s = 0.f;
  for (int j = 0; j < dd; ++j) s += p[j] * p[j];
  sq[n] = s;
}

// Fused decoder: per 16x16 tile compute G=emb@embT and S=scal@scalT via f32 WMMA,
// then out = 1 + tanh(-(sq_i+sq_j-2G)*S), non-temporal stored (10^8 outputs, zero reuse).
__global__ void adj_wmma_kernel(const float* __restrict__ emb, const float* __restrict__ scal,
                                const float* __restrict__ sq, float* __restrict__ out,
                                int Nn, int Dd) {
  int wave = (int)(((unsigned)blockIdx.x * blockDim.x + threadIdx.x) >> 5);
  int lane = threadIdx.x & 31;
  int tiles = Nn >> 4;
  int total = tiles * tiles;
  if (wave >= total) return;
  int i0 = (wave / tiles) << 4;
  int j0 = (wave % tiles) << 4;
  int rI = i0 + (lane & 15);
  int rJ = j0 + (lane & 15);
  int kOff = (lane >> 4) << 1;
  v8f g = {}, s = {};
  for (int k = 0; k < Dd; k += 4) {
    v2f aE, bE, aS, bS;
    const float* p;
    p = emb + (size_t)rI * Dd + k + kOff;  aE.x = p[0]; aE.y = p[1];
    p = emb + (size_t)rJ * Dd + k + kOff;  bE.x = p[0]; bE.y = p[1];  // B = A^T: same frag layout
    p = scal + (size_t)rI * Dd + k + kOff; aS.x = p[0]; aS.y = p[1];
    p = scal + (size_t)rJ * Dd + k + kOff; bS.x = p[0]; bS.y = p[1];
    g = __builtin_amdgcn_wmma_f32_16x16x4_f32(false, aE, false, bE, (short)0, g, false, false);
    s = __builtin_amdgcn_wmma_f32_16x16x4_f32(false, aS, false, bS, (short)0, s, false, false);
  }
  float sqj = sq[rJ];  // column index for this lane
  int mBase = i0 + ((lane >> 4) << 3);
  int nCol = j0 + (lane & 15);
#pragma unroll
  for (int r = 0; r < 8; ++r) {
    int m = mBase + r;
    float dist = -(sq[m] + sqj - 2.f * g[r]);
    float v = 1.f + tanhf(dist * s[r]);
    __builtin_nontemporal_store(v, &out[(size_t)m * Nn + nCol]);
  }
}

extern "C" void kernel_launch(void* const* d_in, const int* in_sizes, int n_in,
                              void* d_out, int out_size, void* d_ws, size_t ws_size,
                              hipStream_t stream) {
  const int T = 8, N = 10000, E = 320000, F = 128, H = 4;
  const float* feat  = (const float*)d_in[0];
  const float* ew    = (const float*)d_in[1];
  const float* Dw    = (const float*)d_in[2];
  const float* W1    = (const float*)d_in[3];
  const float* as1   = (const float*)d_in[4];
  const float* ad1   = (const float*)d_in[5];
  const float* W2    = (const float*)d_in[6];
  const float* as2   = (const float*)d_in[7];
  const float* ad2   = (const float*)d_in[8];
  const float* Wx    = (const float*)d_in[9];
  const float* Wh    = (const float*)d_in[10];
  const float* bl    = (const float*)d_in[11];
  const float* embW  = (const float*)d_in[12];
  const float* embB  = (const float*)d_in[13];
  const float* scW   = (const float*)d_in[14];
  const float* scB   = (const float*)d_in[15];
  const int*   eidx  = (const int*)d_in[16];
  const int*   partA = (const int*)d_in[17];
  float* out = (float*)d_out;
  (void)in_sizes; (void)n_in; (void)out_size; (void)ws_size;

  float* w = (float*)d_ws;
  size_t off = 0;
  auto alloc = [&](size_t nf) { float* p = w + off; off += (nf + 63) & ~(size_t)63; return p; };
  float* hW     = alloc((size_t)N * 128);
  float* alS    = alloc((size_t)N * H);
  float* alD    = alloc((size_t)N * H);
  float* mbuf   = alloc((size_t)N * H);
  float* den    = alloc((size_t)N * H);
  float* num    = alloc((size_t)N * 128);
  float* g1     = alloc((size_t)T * N * 128);  // GAT1 out; later overlaid with agg
  float* g2     = alloc((size_t)T * N * 64);
  float* zbuf   = alloc((size_t)N * 512);
  float* hbuf   = alloc((size_t)N * 128);
  float* cbuf   = alloc((size_t)N * 128);
  float* emb    = alloc((size_t)N * 64);
  float* scal   = alloc((size_t)N * 64);
  float* sq     = alloc((size_t)N);
  float* msum   = alloc(100 * 64);
  float* mden   = alloc(100);
  float* colacc = alloc(64);

  auto gemm = [&](const float* A, const float* B, const float* bias, float* C,
                  int M, int K, int Nc, int acc) {
    int tiles = (M / 16) * (Nc / 16);
    gemm_wmma_kernel<<<cdiv(tiles, 8), TPB, 0, stream>>>(A, B, bias, C, M, K, Nc, acc);
  };

  auto run_gat = [&](const float* X, const float* Wm, const float* a_s, const float* a_d,
                     int dout, int dh, const int* srcT, const int* dstT, const float* ewT,
                     float* outp) {
    gemm(X, Wm, nullptr, hW, N, 128, dout, 0);
    node_alpha_kernel<<<cdiv((long long)N * H, TPB), TPB, 0, stream>>>(hW, a_s, a_d, alS, alD, N, H, dh);
    fill_kernel<<<cdiv((long long)N * H, TPB), TPB, 0, stream>>>(mbuf, -INFINITY, N * H);
    fill_kernel<<<cdiv((long long)N * H, TPB), TPB, 0, stream>>>(den, 0.f, N * H);
    fill_kernel<<<cdiv((long long)N * dout, TPB), TPB, 0, stream>>>(num, 0.f, N * dout);
    edge_max_kernel<<<cdiv((long long)E * H, TPB), TPB, 0, stream>>>(srcT, dstT, alS, alD, mbuf, E, H);
    fix_m_kernel<<<cdiv((long long)N * H, TPB), TPB, 0, stream>>>(mbuf, N * H);
    edge_accum_kernel<<<cdiv((long long)E * H, TPB), TPB, 0, stream>>>(srcT, dstT, alS, alD, mbuf,
                                                                       ewT, hW, den, num, E, H, dh);
    gat_out_kernel<<<cdiv((long long)N * dout, TPB), TPB, 0, stream>>>(num, den, outp, N, H, dh);
  };

  // ---- GAT layer 1 (all timesteps) ----
  for (int t = 0; t < T; ++t) {
    const int* srcT = eidx + (size_t)t * 2 * E;
    const int* dstT = srcT + E;
    run_gat(feat + (size_t)t * N * F, W1, as1, ad1, 128, 32, srcT, dstT,
            ew + (size_t)t * E, g1 + (size_t)t * N * 128);
  }

  // ---- GAT layer 2 + meso + agg (agg overlays g1[t] after it is consumed) ----
  for (int t = 0; t < T; ++t) {
    const int* srcT = eidx + (size_t)t * 2 * E;
    const int* dstT = srcT + E;
    float* g2t = g2 + (size_t)t * N * 64;
    run_gat(g1 + (size_t)t * N * 128, W2, as2, ad2, 64, 16, srcT, dstT,
            ew + (size_t)t * E, g2t);
    const int* pt = partA + (size_t)t * N;
    fill_kernel<<<cdiv(6400, TPB), TPB, 0, stream>>>(msum, 0.f, 6400);
    fill_kernel<<<1, TPB, 0, stream>>>(mden, 0.f, 100);
    meso_kernel<<<cdiv((long long)N * 64, TPB), TPB, 0, stream>>>(g2t, pt, Dw + (size_t)t * N,
                                                                  msum, mden, N, 64);
    agg_kernel<<<cdiv((long long)N * 64, TPB), TPB, 0, stream>>>(g1 + (size_t)t * N * 128, g2t,
                                                                 msum, mden, pt, N, 64);
  }

  // ---- LSTM over T steps ----
  fill_kernel<<<cdiv((long long)N * 128, TPB), TPB, 0, stream>>>(hbuf, 0.f, N * 128);
  fill_kernel<<<cdiv((long long)N * 128, TPB), TPB, 0, stream>>>(cbuf, 0.f, N * 128);
  for (int t = 0; t < T; ++t) {
    const float* aggT = g1 + (size_t)t * N * 128;
    gemm(aggT, Wx, bl, zbuf, N, 128, 512, 0);
    gemm(hbuf, Wh, nullptr, zbuf, N, 128, 512, 1);
    lstm_kernel<<<cdiv((long long)N * 128, TPB), TPB, 0, stream>>>(zbuf, hbuf, cbuf, N, 128);
  }

  // ---- Decoder ----
  gemm(hbuf, embW, embB, emb, N, 128, 64, 0);
  tanh_kernel<<<cdiv((long long)N * 64, TPB), TPB, 0, stream>>>(emb, N * 64);
  fill_kernel<<<1, TPB, 0, stream>>>(colacc, 0.f, 64);
  colsq_kernel<<<cdiv((long long)N * 64, TPB), TPB, 0, stream>>>(emb, colacc, N * 64, 64);
  colscale_kernel<<<cdiv((long long)N * 64, TPB), TPB, 0, stream>>>(emb, colacc, N * 64, 64);
  gemm(hbuf, scW, scB, scal, N, 128, 64, 0);
  sigmoid_kernel<<<cdiv((long long)N * 64, TPB), TPB, 0, stream>>>(scal, N * 64);
  sqnorm_kernel<<<cdiv(N, TPB), TPB, 0, stream>>>(emb, sq, N, 64);

  // ---- Fused pairwise output: 625x625 tiles, 8 waves/block ----
  {
    long long tiles = (long long)(N / 16) * (N / 16);
    adj_wmma_kernel<<<cdiv(tiles, 8), TPB, 0, stream>>>(emb, scal, sq, out, N, 64);
  }
}